// Contexture_43946105373418
// MI455X (gfx1250) — compile-verified
//
#include <hip/hip_runtime.h>

#define P_    6
#define B_    8
#define C_    10
#define H_    192
#define W_    192
#define K2_   9
#define OCH_  27
#define KDIM_ 90
#define KPAD_ 92
#define HW_   (H_*W_)
#define CHW_  (C_*HW_)
#define NTOT_ (P_*B_*CHW_)
#define STRIPS_X (W_/16)
#define STRIPS_IMG (H_*STRIPS_X)   // 2304 strips per (p,b) image, multiple of WAVES_
#define WAVES_ 4
#define NTHR_  (WAVES_*32)
#define EPS_  1e-5f

typedef __attribute__((ext_vector_type(2))) float v2f;
typedef __attribute__((ext_vector_type(8))) float v8f;

// Branch-free im2col sample: clamp to a safe address, select zero when OOB.
__device__ __forceinline__ float sample_tap(const float* __restrict__ img,
                                            int2 t, int y, int px) {
  int kyo = t.y >> 16;                     // ky*dil (sentinel 16384 for pad rows)
  int kxo = (int)(short)(t.y & 0xffff);    // kx*dil
  int yy = y + kyo, xx = px + kxo;
  bool v = ((unsigned)yy < (unsigned)H_) && ((unsigned)xx < (unsigned)W_);
  int cy = min(max(yy, 0), H_ - 1);
  int cx = min(max(xx, 0), W_ - 1);
  float val = img[t.x + cy * W_ + cx];     // always in-range -> unconditional load
  return v ? val : 0.f;                    // v_cndmask, no exec churn
}

// One deformable-conv + BN + ReLU + residual stage. If att != nullptr, the
// final attention factor (1 - att/max) is fused into the epilogue (stage 2).
// One wave handles a 16-pixel row strip. Wave32, EXEC all-1s at every WMMA.
__global__ __launch_bounds__(NTHR_) void dcn_stage_kernel(
    const float* __restrict__ hin, float* __restrict__ hout,
    const float* __restrict__ ow,  const float* __restrict__ ob,
    const float* __restrict__ cw,
    const float* __restrict__ bng, const float* __restrict__ bnb,
    const float* __restrict__ bnm, const float* __restrict__ bnv,
    const float* __restrict__ att, const unsigned* __restrict__ maxbits,
    int stage, int dil)
{
  __shared__ float s_owt[32 * KPAD_];          // offset weights, zero-padded [m][92]
  __shared__ float s_cwt[16 * KPAD_];          // main-conv weights, zero-padded [m][92]
  __shared__ int2  s_tab[KPAD_];               // per-gk tap table (c*HW, pack(kyo,kxo))
  __shared__ float s_om [WAVES_][OCH_ * 16];   // offset-conv output (27 ch x 16 px)
  __shared__ float s_val[WAVES_][16 * KPAD_];  // masked im2col values [pixel][92]

  const int tid  = threadIdx.x;
  const int wave = tid >> 5;
  const int lane = tid & 31;
  const int n16  = lane & 15;
  const int hi   = lane >> 4;

  // Block-uniform (p,b): STRIPS_IMG % WAVES_ == 0.
  const int pb = (blockIdx.x * WAVES_) / STRIPS_IMG;      // p*B_ + b
  const int p  = pb / B_;

  int gstrip = blockIdx.x * WAVES_ + wave;
  int sloc   = gstrip % STRIPS_IMG;            // strip within image
  int sx = sloc % STRIPS_X;
  int y  = sloc / STRIPS_X;
  int x0 = sx * 16;
  int px = x0 + n16;

  const float* img  = hin  + (size_t)pb * CHW_;
  float*       oimg = hout + (size_t)pb * CHW_;
  const float* owp  = ow  + (size_t)(p * 3 + stage) * (OCH_ * KDIM_);
  const float* obp  = ob  + (p * 3 + stage) * OCH_;
  const float* cwp  = cw  + (size_t)(p * 3 + stage) * (C_ * KDIM_);
  const float* gp   = bng + (p * 3 + stage) * C_;
  const float* btp  = bnb + (p * 3 + stage) * C_;
  const float* mnp  = bnm + (p * 3 + stage) * C_;
  const float* vrp  = bnv + (p * 3 + stage) * C_;

  __builtin_prefetch(owp, 0, 1);
  __builtin_prefetch(cwp, 0, 1);

  // ---- Stage weights (zero-padded) + tap table into LDS, once per block ----
  for (int i = tid; i < 32 * KPAD_; i += NTHR_) {
    int m = i / KPAD_, gk = i - m * KPAD_;
    s_owt[i] = (m < OCH_ && gk < KDIM_) ? owp[m * KDIM_ + gk] : 0.f;
  }
  for (int i = tid; i < 16 * KPAD_; i += NTHR_) {
    int m = i / KPAD_, gk = i - m * KPAD_;
    s_cwt[i] = (m < C_ && gk < KDIM_) ? cwp[m * KDIM_ + gk] : 0.f;
  }
  if (tid < KPAD_) {
    int gk = tid;
    int2 e;
    if (gk < KDIM_) {
      int c = gk / K2_, t = gk - c * K2_;
      int kyo = (t / 3 - 1) * dil;
      int kxo = (t % 3 - 1) * dil;
      e.x = c * HW_;
      e.y = (kyo << 16) | (kxo & 0xffff);
    } else {                                   // K-pad rows: always-invalid sentinel
      e.x = 0;
      e.y = (16384 << 16);
    }
    s_tab[gk] = e;
  }
  // Zero K-padding columns 90..91 of this wave's value tile (32 == lane count).
  s_val[wave][n16 * KPAD_ + KDIM_ + hi] = 0.f;
  __syncthreads();

  // ---- Phase 1: offset conv, (27x90)@(90x16) via V_WMMA_F32_16X16X4_F32 ----
  v8f acc0 = {0.f,0.f,0.f,0.f,0.f,0.f,0.f,0.f};
  v8f acc1 = acc0;
  for (int kk = 0; kk < KPAD_; kk += 4) {
    int gkb = kk + hi * 2;                     // even -> 8B-aligned LDS pairs
    v2f a0 = *(const v2f*)&s_owt[n16 * KPAD_ + gkb];
    v2f a1 = *(const v2f*)&s_owt[(n16 + 16) * KPAD_ + gkb];
    int2 t0 = s_tab[gkb];
    int2 t1 = s_tab[gkb + 1];
    v2f bf;
    bf.x = sample_tap(img, t0, y, px);
    bf.y = sample_tap(img, t1, y, px);
    acc0 = __builtin_amdgcn_wmma_f32_16x16x4_f32(false, a0, false, bf, (short)0, acc0, false, false);
    acc1 = __builtin_amdgcn_wmma_f32_16x16x4_f32(false, a1, false, bf, (short)0, acc1, false, false);
  }
  // D layout: VGPR r holds channel (r + 8*hi), column n16. Scatter + bias to LDS.
  #pragma unroll
  for (int r = 0; r < 8; ++r) {
    int ch0 = r + hi * 8;                      // 0..15
    s_om[wave][ch0 * 16 + n16] = acc0[r] + obp[ch0];
    int ch1 = 16 + r + hi * 8;                 // 16..31 (valid < 27)
    if (ch1 < OCH_) s_om[wave][ch1 * 16 + n16] = acc1[r] + obp[ch1];
  }
  __syncthreads();

  // ---- Phase 2: deformable bilinear sampling -> masked im2col in LDS ----
  for (int it = 0; it < 5; ++it) {
    int task = it * 32 + lane;                 // 144 = 16 px * 9 taps
    if (task < 16 * K2_) {
      int n = task & 15;
      int k = task >> 4;
      float offy = s_om[wave][(2 * k + 0) * 16 + n];
      float offx = s_om[wave][(2 * k + 1) * 16 + n];
      float mraw = s_om[wave][(18 + k) * 16 + n];
      float msk  = 1.f / (1.f + expf(-mraw));
      int kyi = k / 3 - 1, kxi = k % 3 - 1;
      float py  = (float)y        + (float)(kyi * dil) + offy;
      float pxf = (float)(x0 + n) + (float)(kxi * dil) + offx;
      float fy0 = floorf(py), fx0 = floorf(pxf);
      float wy = py - fy0, wx = pxf - fx0;
      int iy0 = (int)fy0, ix0 = (int)fx0;
      int iy1 = iy0 + 1,  ix1 = ix0 + 1;
      bool vy0 = (iy0 >= 0) && (iy0 < H_), vy1 = (iy1 >= 0) && (iy1 < H_);
      bool vx0 = (ix0 >= 0) && (ix0 < W_), vx1 = (ix1 >= 0) && (ix1 < W_);
      int cy0 = min(max(iy0, 0), H_ - 1), cy1 = min(max(iy1, 0), H_ - 1);
      int cx0 = min(max(ix0, 0), W_ - 1), cx1 = min(max(ix1, 0), W_ - 1);
      int a00 = cy0 * W_ + cx0, a01 = cy0 * W_ + cx1;
      int a10 = cy1 * W_ + cx0, a11 = cy1 * W_ + cx1;
      float w00 = (1.f - wy) * (1.f - wx) * (float)(vy0 && vx0);
      float w01 = (1.f - wy) * wx         * (float)(vy0 && vx1);
      float w10 = wy * (1.f - wx)         * (float)(vy1 && vx0);
      float w11 = wy * wx                 * (float)(vy1 && vx1);
      #pragma unroll
      for (int c = 0; c < C_; ++c) {
        const float* cb = img + c * HW_;
        float v = w00 * cb[a00] + w01 * cb[a01] + w10 * cb[a10] + w11 * cb[a11];
        s_val[wave][n * KPAD_ + c * K2_ + k] = v * msk;
      }
    }
  }
  __syncthreads();

  // ---- Phase 3: main conv, (10x90)@(90x16) via WMMA, then BN+ReLU+residual ----
  v8f acc = {0.f,0.f,0.f,0.f,0.f,0.f,0.f,0.f};
  for (int kk = 0; kk < KPAD_; kk += 4) {
    int gkb = kk + hi * 2;
    v2f a  = *(const v2f*)&s_cwt[n16 * KPAD_ + gkb];
    v2f bf = *(const v2f*)&s_val[wave][n16 * KPAD_ + gkb];
    acc = __builtin_amdgcn_wmma_f32_16x16x4_f32(false, a, false, bf, (short)0, acc, false, false);
  }
  // Fused attention factor for the last stage: (1 - sumsq/max), uniform per pixel.
  float fac = 1.f;
  if (att) {
    float mx = __uint_as_float(maxbits[p]);
    fac = 1.f - att[(size_t)pb * HW_ + y * W_ + px] / mx;
  }
  #pragma unroll
  for (int r = 0; r < 8; ++r) {
    int ch = r + hi * 8;
    if (ch < C_) {
      float sc = gp[ch] / sqrtf(vrp[ch] + EPS_);
      float sh = btp[ch] - mnp[ch] * sc;
      float o  = fmaxf(acc[r] * sc + sh, 0.f);
      int idx  = ch * HW_ + y * W_ + px;
      oimg[idx] = (o + img[idx]) * fac;
    }
  }
}

// Per-pixel sum of squares over channels + per-part running max (uint trick, vals >= 0).
__global__ __launch_bounds__(256) void att_kernel(const float* __restrict__ xp,
                                                  float* __restrict__ att,
                                                  unsigned* __restrict__ maxbits)
{
  __shared__ float red[256];
  int idx = blockIdx.x * 256 + threadIdx.x;    // over P*B*HW (exact multiple of 256)
  int p   = idx / (B_ * HW_);                  // uniform within a block
  int rem = idx - p * (B_ * HW_);
  int b   = rem / HW_;
  int hw  = rem - b * HW_;
  const float* base = xp + (size_t)(p * B_ + b) * CHW_ + hw;
  float s = 0.f;
  #pragma unroll
  for (int c = 0; c < C_; ++c) { float v = base[c * HW_]; s += v * v; }
  att[idx] = s;
  red[threadIdx.x] = s;
  __syncthreads();
  for (int off = 128; off > 0; off >>= 1) {
    if (threadIdx.x < off)
      red[threadIdx.x] = fmaxf(red[threadIdx.x], red[threadIdx.x + off]);
    __syncthreads();
  }
  if (threadIdx.x == 0) atomicMax(&maxbits[p], __float_as_uint(red[0]));
}

extern "C" void kernel_launch(void* const* d_in, const int* in_sizes, int n_in,
                              void* d_out, int out_size, void* d_ws, size_t ws_size,
                              hipStream_t stream)
{
  const float* xp = (const float*)d_in[0];
  const float* ow = (const float*)d_in[1];
  const float* ob = (const float*)d_in[2];
  const float* cw = (const float*)d_in[3];
  const float* g  = (const float*)d_in[4];
  const float* bt = (const float*)d_in[5];
  const float* mn = (const float*)d_in[6];
  const float* vr = (const float*)d_in[7];
  float* outf = (float*)d_out;

  // Workspace: h ping-pong buffer (d_out is the other one) + att sums + per-part max.
  float*    ws_h    = (float*)d_ws;
  float*    att     = ws_h + (size_t)NTOT_;
  unsigned* maxbits = (unsigned*)(att + (size_t)P_ * B_ * HW_);

  hipMemsetAsync(maxbits, 0, P_ * sizeof(unsigned), stream);
  att_kernel<<<(P_ * B_ * HW_) / 256, 256, 0, stream>>>(xp, att, maxbits);

  int nstrips = P_ * B_ * STRIPS_IMG;          // 110592, exact multiple of WAVES_
  dim3 grid(nstrips / WAVES_), block(NTHR_);
  // Stage 0: xp -> d_out ; Stage 1: d_out -> ws_h ; Stage 2 (+att): ws_h -> d_out
  dcn_stage_kernel<<<grid, block, 0, stream>>>(xp,   outf, ow, ob, cw, g, bt, mn, vr,
                                               nullptr, nullptr, 0, 2);
  dcn_stage_kernel<<<grid, block, 0, stream>>>(outf, ws_h, ow, ob, cw, g, bt, mn, vr,
                                               nullptr, nullptr, 1, 4);
  dcn_stage_kernel<<<grid, block, 0, stream>>>(ws_h, outf, ow, ob, cw, g, bt, mn, vr,
                                               att, maxbits, 2, 8);
}